// CompleteSpatialConvolution_86878598463516
// MI455X (gfx1250) — compile-verified
//
#include <hip/hip_runtime.h>
#include <math.h>

#define NN     384
#define DFEAT  480
#define K0     9
#define K1     20
#define K2     22
#define FROW   489      // 129 + 65*3 + 33*5
#define TSCAL  197      // scal row stride (193 valid + 3 zero pad + 1)

#define W1P_ELEMS (49*2*32*2)    // padded/swizzled W_m1 for GEMM1 B-operand
#define W2P_ELEMS (8*2*240*2)    // padded/swizzled W_m2 for GEMM2 B-operand

typedef __attribute__((ext_vector_type(2))) float v2f;
typedef __attribute__((ext_vector_type(8))) float v8f;

// ---------------------------------------------------------------------------
// f32 WMMA 16x16x4 wrapper (D = A[16x4] * B[4x16] + C[16x16], all fp32)
// ---------------------------------------------------------------------------
__device__ __forceinline__ v8f wmma4(v2f a, v2f b, v8f c) {
  return __builtin_amdgcn_wmma_f32_16x16x4_f32(false, a, false, b, (short)0, c,
                                               false, false);
}

// ---------------------------------------------------------------------------
// Kernel 1: Clebsch-Gordan tensors (exact mirror of the reference formula)
// ---------------------------------------------------------------------------
__device__ const int PATHTAB[15][4] = {
  // l3, l1, l2, kbase
  {0,0,0,0},{0,1,1,1},{0,2,2,4},
  {1,0,1,0},{1,1,0,3},{1,1,1,4},{1,1,2,7},{1,2,1,12},{1,2,2,15},
  {2,0,2,0},{2,1,1,5},{2,1,2,8},{2,2,0,13},{2,2,1,14},{2,2,2,17}};

__device__ double dfact_(int n){ double r=1.0; for(int i=2;i<=n;++i) r*=(double)i; return r; }

__device__ double cgco_(int j1,int m1,int j2,int m2,int j3,int m3){
  if (m1 + m2 != m3) return 0.0;
  double pref = sqrt((double)(2*j3+1) * dfact_(j3+j1-j2)*dfact_(j3-j1+j2)*dfact_(j1+j2-j3)/dfact_(j1+j2+j3+1));
  pref *= sqrt(dfact_(j3+m3)*dfact_(j3-m3)*dfact_(j1-m1)*dfact_(j1+m1)*dfact_(j2-m2)*dfact_(j2+m2));
  double s = 0.0;
  for (int k = 0; k <= j1+j2+j3; ++k){
    int d0=k, d1=j1+j2-j3-k, d2=j1-m1-k, d3=j2+m2-k, d4=j3-j2+m1+k, d5=j3-j1-m2+k;
    if (d0<0||d1<0||d2<0||d3<0||d4<0||d5<0) continue;
    double den = dfact_(d0)*dfact_(d1)*dfact_(d2)*dfact_(d3)*dfact_(d4)*dfact_(d5);
    s += ((k & 1) ? -1.0 : 1.0) / den;
  }
  return pref * s;
}

__device__ void u_real_(int l, double Ur[5][5], double Ui[5][5]){
  for (int a=0;a<5;++a) for (int b=0;b<5;++b){ Ur[a][b]=0.0; Ui[a][b]=0.0; }
  const double s2 = 0.70710678118654752440;
  for (int m=-l; m<=l; ++m){
    int row = m + l;
    if (m == 0) { Ur[l][l] = 1.0; }
    else if (m > 0){
      double sg = (m & 1) ? -1.0 : 1.0;        // (-1)^m
      Ur[row][ m+l] = sg * s2;
      Ur[row][-m+l] = s2;
    } else {
      int ma = -m; double sg = (ma & 1) ? -1.0 : 1.0;  // (-1)^m for m<0
      Ui[row][ m+l] = s2;
      Ui[row][-m+l] = -sg * s2;
    }
  }
}

__global__ void eq_init_cg(float* __restrict__ CGP0, float* __restrict__ CGP1,
                           float* __restrict__ CGP2){
  int t = threadIdx.x;
  if (t >= 15) return;
  int l3 = PATHTAB[t][0], l1 = PATHTAB[t][1], l2 = PATHTAB[t][2], kb = PATHTAB[t][3];
  int d1 = 2*l1+1, d2 = 2*l2+1, d3 = 2*l3+1;

  double C[5][5][5];
  for (int a=0;a<5;++a) for(int b=0;b<5;++b) for(int c=0;c<5;++c) C[a][b][c]=0.0;
  for (int m1=-l1; m1<=l1; ++m1)
    for (int m2=-l2; m2<=l2; ++m2){
      int m3 = m1 + m2;
      if (m3 >= -l3 && m3 <= l3) C[m1+l1][m2+l2][m3+l3] = cgco_(l1,m1,l2,m2,l3,m3);
    }

  double U1r[5][5],U1i[5][5],U2r[5][5],U2i[5][5],U3r[5][5],U3i[5][5];
  u_real_(l1,U1r,U1i); u_real_(l2,U2r,U2i); u_real_(l3,U3r,U3i);

  double Rre[5][5][5], Rim[5][5][5];
  double mxr = 0.0, mxi = 0.0;
  for (int i=0;i<d1;++i) for (int j=0;j<d2;++j) for (int k=0;k<d3;++k){
    double sre=0.0, sim=0.0;
    for (int a=0;a<d1;++a) for (int b=0;b<d2;++b){
      double pre = U1r[i][a]*U2r[j][b] - U1i[i][a]*U2i[j][b];
      double pim = U1r[i][a]*U2i[j][b] + U1i[i][a]*U2r[j][b];
      for (int c=0;c<d3;++c){
        double cv = C[a][b][c];
        if (cv == 0.0) continue;
        double qre = pre*U3r[k][c] + pim*U3i[k][c];   // * conj(U3[k][c])
        double qim = pim*U3r[k][c] - pre*U3i[k][c];
        sre += qre * cv; sim += qim * cv;
      }
    }
    Rre[i][j][k]=sre; Rim[i][j][k]=sim;
    double ar = fabs(sre), ai = fabs(sim);
    if (ar > mxr) mxr = ar;
    if (ai > mxi) mxi = ai;
  }
  int useRe = (mxr >= mxi);

  for (int b=0;b<d2;++b){
    int k = kb + b;
    for (int a=0;a<5;++a)
      for (int c=0;c<d3;++c){
        double v = 0.0;
        if (a < d1) v = useRe ? Rre[a][b][c] : Rim[a][b][c];
        if      (l3 == 0) CGP0[k*5 + a]         = (float)v;
        else if (l3 == 1) CGP1[(k*5 + a)*3 + c] = (float)v;
        else              CGP2[(k*5 + a)*5 + c] = (float)v;
      }
  }
}

// ---------------------------------------------------------------------------
// Kernel 1b: pack MLP weights into padded, WMMA-B-operand-swizzled layout.
// W1p[s][e2][n][e1] = W_m1[4s+2*e2+e1][n]  (zero for k>=193), s<49, n<32
// W2p[s][e2][n][e1] = W_m2[4s+2*e2+e1][n]  (zero for n>=227), s<8,  n<240
// Each lane's two K-values become one aligned 8-byte load (global_load_b64).
// ---------------------------------------------------------------------------
__global__ __launch_bounds__(256) void eq_pack_w(
    const float* __restrict__ Wm1, const float* __restrict__ Wm2,
    float* __restrict__ W1p, float* __restrict__ W2p)
{
  int tid = blockIdx.x * blockDim.x + threadIdx.x;
  int stride = gridDim.x * blockDim.x;
  for (int idx = tid; idx < W1P_ELEMS; idx += stride){
    int e1 = idx & 1;
    int n  = (idx >> 1) & 31;
    int se = idx >> 6;              // s*2 + e2
    int k  = 2*se + e1;             // = 4s + 2*e2 + e1
    W1p[idx] = (k < 193) ? Wm1[k*32 + n] : 0.f;
  }
  for (int idx = tid; idx < W2P_ELEMS; idx += stride){
    int e1 = idx & 1;
    int n  = (idx >> 1) % 240;
    int se = (idx >> 1) / 240;      // s*2 + e2
    int k  = 2*se + e1;
    W2p[idx] = (n < 227) ? Wm2[k*227 + n] : 0.f;
  }
}

// ---------------------------------------------------------------------------
// Kernel 2: per-node path-weighted features T[l3][j][k][v]
// ---------------------------------------------------------------------------
__global__ __launch_bounds__(128) void eq_stage_t(
    const float* __restrict__ x,
    const float* __restrict__ W0, const float* __restrict__ W1w, const float* __restrict__ W2w,
    float* __restrict__ T0, float* __restrict__ T1, float* __restrict__ T2)
{
  const int j = blockIdx.x;
  const int tid = threadIdx.x;
  __shared__ float xs[DFEAT];
  for (int q = tid; q < DFEAT; q += 128) xs[q] = x[(size_t)j*DFEAT + q];
  __syncthreads();

  { // T0: v = tid (0..127); k-order: (0,0), (1,1)b, (2,2)b
    const int v = tid;
    float* T0j = T0 + (size_t)j*K0*128;
    float acc = 0.f;
    for (int u=0;u<128;++u) acc += xs[u] * W0[u*128 + v];
    T0j[v] = acc;
    for (int b=0;b<3;++b){
      acc = 0.f;
      for (int u=0;u<64;++u) acc += xs[128+u*3+b] * W0[(128+u)*128 + v];
      T0j[(1+b)*128 + v] = acc;
    }
    for (int b=0;b<5;++b){
      acc = 0.f;
      for (int u=0;u<32;++u) acc += xs[320+u*5+b] * W0[(192+u)*128 + v];
      T0j[(4+b)*128 + v] = acc;
    }
  }
  if (tid < 64){ // T1: k-order: (0,1),(1,0),(1,1),(1,2),(2,1),(2,2)
    const int v = tid;
    float* T1j = T1 + (size_t)j*K1*64;
    int k = 0; float acc;
    for (int b=0;b<3;++b){ acc=0.f; for(int u=0;u<64;++u)  acc += xs[128+u*3+b]*W1w[u*64+v];        T1j[k*64+v]=acc; ++k; }
    { acc=0.f;             for(int u=0;u<128;++u) acc += xs[u]        *W1w[(64+u)*64+v];            T1j[k*64+v]=acc; ++k; }
    for (int b=0;b<3;++b){ acc=0.f; for(int u=0;u<64;++u)  acc += xs[128+u*3+b]*W1w[(192+u)*64+v];  T1j[k*64+v]=acc; ++k; }
    for (int b=0;b<5;++b){ acc=0.f; for(int u=0;u<32;++u)  acc += xs[320+u*5+b]*W1w[(256+u)*64+v];  T1j[k*64+v]=acc; ++k; }
    for (int b=0;b<3;++b){ acc=0.f; for(int u=0;u<64;++u)  acc += xs[128+u*3+b]*W1w[(288+u)*64+v];  T1j[k*64+v]=acc; ++k; }
    for (int b=0;b<5;++b){ acc=0.f; for(int u=0;u<32;++u)  acc += xs[320+u*5+b]*W1w[(352+u)*64+v];  T1j[k*64+v]=acc; ++k; }
  } else if (tid < 96){ // T2: k-order: (0,2),(1,1),(1,2),(2,0),(2,1),(2,2)
    const int v = tid - 64;
    float* T2j = T2 + (size_t)j*K2*32;
    int k = 0; float acc;
    for (int b=0;b<5;++b){ acc=0.f; for(int u=0;u<32;++u)  acc += xs[320+u*5+b]*W2w[u*32+v];        T2j[k*32+v]=acc; ++k; }
    for (int b=0;b<3;++b){ acc=0.f; for(int u=0;u<64;++u)  acc += xs[128+u*3+b]*W2w[(32+u)*32+v];   T2j[k*32+v]=acc; ++k; }
    for (int b=0;b<5;++b){ acc=0.f; for(int u=0;u<32;++u)  acc += xs[320+u*5+b]*W2w[(96+u)*32+v];   T2j[k*32+v]=acc; ++k; }
    { acc=0.f;             for(int u=0;u<128;++u) acc += xs[u]        *W2w[(128+u)*32+v];           T2j[k*32+v]=acc; ++k; }
    for (int b=0;b<3;++b){ acc=0.f; for(int u=0;u<64;++u)  acc += xs[128+u*3+b]*W2w[(256+u)*32+v];  T2j[k*32+v]=acc; ++k; }
    for (int b=0;b<5;++b){ acc=0.f; for(int u=0;u<32;++u)  acc += xs[320+u*5+b]*W2w[(320+u)*32+v];  T2j[k*32+v]=acc; ++k; }
  }
}

// ---------------------------------------------------------------------------
// Kernel 3: fused pairwise messages + WMMA MLP gate + gated aggregation
// ---------------------------------------------------------------------------
__device__ const int L1K0[K0]  = {0,1,1,1,2,2,2,2,2};
__device__ const int L1K1[K1]  = {0,0,0,1,1,1,1,1,1,1,1,1,2,2,2,2,2,2,2,2};
__device__ const int L1K2[K2]  = {0,0,0,0,0,1,1,1,1,1,1,1,1,2,2,2,2,2,2,2,2,2};

__device__ __forceinline__ float angdotS(int l1, const float* ang, float cm,
                                         const float* cg, int stride){
  if (l1 == 0) return cm * cg[0];
  if (l1 == 1) return ang[0]*cg[0] + ang[1]*cg[stride] + ang[2]*cg[2*stride];
  return ang[3]*cg[0] + ang[4]*cg[stride] + ang[5]*cg[2*stride]
       + ang[6]*cg[3*stride] + ang[7]*cg[4*stride];
}

__global__ __launch_bounds__(32) void eq_pair_kernel(
    const float* __restrict__ coord, const unsigned char* __restrict__ mask,
    const float* __restrict__ emb,
    const float* __restrict__ bm1, const float* __restrict__ bm2,
    const float* __restrict__ W1p, const float* __restrict__ W2p,
    const float* __restrict__ T0, const float* __restrict__ T1g, const float* __restrict__ T2g,
    const float* __restrict__ CGP0, const float* __restrict__ CGP1, const float* __restrict__ CGP2,
    float* __restrict__ fglob, int jchunk)
{
  __shared__ float scal[16][TSCAL];   // [pair][193-col scal, zero-padded to 196]
  __shared__ float B1s[16][K1][3];    // per-pair angular contraction for l3=1
  __shared__ float B2s[16][K2][5];    // per-pair angular contraction for l3=2
  __shared__ float angs[16][10];      // ang[0..7], cm at [8]
  __shared__ float hls[16][33];       // silu(hidden) for GEMM2 A-operand

  const int lane = threadIdx.x;
  const int ii   = lane & 15;
  const int half = lane >> 4;
  const int kadd = 2 * half;
  const int i0   = blockIdx.x * 16;
  const int i    = i0 + ii;

  const float s3  = 1.7320508075688772f;
  const float s5  = 2.2360679774997896f;
  const float s15 = 3.8729833462074170f;

  const v2f* __restrict__ W1v = (const v2f*)W1p;   // [ (s*2+half)*32 + n ]
  const v2f* __restrict__ W2v = (const v2f*)W2p;   // [ (s*2+half)*240 + n ]

  const float ci0 = coord[3*i+0], ci1 = coord[3*i+1], ci2 = coord[3*i+2];
  const int   mi  = (mask[i] != 0);

  if (half == 0){ scal[ii][193] = 0.f; scal[ii][194] = 0.f; scal[ii][195] = 0.f; }

  const int jbeg = blockIdx.y * jchunk;
  for (int j = jbeg; j < jbeg + jchunk; ++j){
    // ---- phase 1: geometry, SH, radial, rel-embed, msg0 -> scal row -------
    const int mj = (mask[j] != 0);
    const float cm = (mi && mj) ? 1.f : 0.f;
    const float vx = (ci0 - coord[3*j+0]) * cm;
    const float vy = (ci1 - coord[3*j+1]) * cm;
    const float vz = (ci2 - coord[3*j+2]) * cm;
    const float nsq = vx*vx + vy*vy + vz*vz;
    const float r = (nsq == 0.f) ? 0.f : sqrtf(nsq);
    float ang[8];
    ang[0] = s3*vy; ang[1] = s3*vz; ang[2] = s3*vx;
    ang[3] = s15*vx*vy; ang[4] = s15*vy*vz;
    ang[5] = 0.5f*s5*(2.f*vz*vz - vx*vx - vy*vy);
    ang[6] = s15*vx*vz; ang[7] = 0.5f*s15*(vx*vx - vy*vy);
    if (half == 0){
      #pragma unroll
      for (int q=0;q<8;++q) angs[ii][q] = ang[q];
      angs[ii][8] = cm;
    }
    int rel = i - j;
    if (rel > 16 || rel < -16) rel = 0;
    if (cm == 0.f) rel = 0;
    rel += 16;
    #pragma unroll
    for (int c=0;c<16;++c) scal[ii][half*16 + c] = emb[rel*32 + half*16 + c];
    const float invstep = 33.f/12.f;
    #pragma unroll
    for (int c=0;c<16;++c){
      int kk = half*16 + c;
      float vv = (float)(kk+1) * (12.f/33.f);
      float z = (r - vv) * invstep;
      scal[ii][32 + kk] = expf(-z*z) * (1.f/1.12f) * cm;
    }
    // angular contraction matrices
    float B0[K0];
    #pragma unroll
    for (int k=0;k<K0;++k) B0[k] = angdotS(L1K0[k], ang, cm, &CGP0[k*5], 1);
    if (half == 0){
      for (int k=0;k<K1;++k)
        #pragma unroll
        for (int c=0;c<3;++c)
          B1s[ii][k][c] = angdotS(L1K1[k], ang, cm, &CGP1[k*15 + c], 3);
    } else {
      for (int k=0;k<K2;++k)
        #pragma unroll
        for (int c=0;c<5;++c)
          B2s[ii][k][c] = angdotS(L1K2[k], ang, cm, &CGP2[k*25 + c], 5);
    }
    // msg0 (l3=0 messages), v-range split across halves
    const float* T0j = T0 + (size_t)j*K0*128;
    for (int v = half*64; v < half*64 + 64; ++v){
      float acc = 0.f;
      #pragma unroll
      for (int k=0;k<K0;++k) acc += B0[k] * T0j[k*128 + v];
      scal[ii][64 + v] = acc;
    }
    if (half == 0) scal[ii][192] = cm;   // M0's last column is A[0] = cm
    __builtin_amdgcn_wave_barrier();

    if (j + 1 < NN) __builtin_prefetch((const void*)(T0 + (size_t)(j+1)*K0*128), 0, 1);

    // ---- GEMM1: [16 x 193] @ W_m1[193 x 32] via f32 WMMA 16x16x4 ----------
    // Branch-free: A from zero-padded LDS row, B as one global_load_b64 from
    // the pre-swizzled padded weight image.
    v8f D0 = {}; v8f D1 = {};
    #pragma unroll 7
    for (int s = 0; s < 49; ++s){
      int kk = 4*s + kadd;
      v2f A; A.x = scal[ii][kk]; A.y = scal[ii][kk+1];
      int bidx = (s*2 + half)*32 + ii;
      v2f Bl = W1v[bidx];
      v2f Bh = W1v[bidx + 16];
      D0 = wmma4(A, Bl, D0);
      D1 = wmma4(A, Bh, D1);
    }
    // bias + silu, scatter to h LDS in A-operand friendly layout
    #pragma unroll
    for (int rg=0; rg<8; ++rg){
      int m = 8*half + rg;
      float x0 = D0[rg] + bm1[ii];
      float x1 = D1[rg] + bm1[16 + ii];
      hls[m][ii]      = x0 / (1.f + expf(-x0));
      hls[m][16 + ii] = x1 / (1.f + expf(-x1));
    }
    __builtin_amdgcn_wave_barrier();

    // ---- GEMM2: h[16 x 32] @ W_m2[32 x 227] + in-place gated consume ------
    const float* T1j = T1g + (size_t)j*K1*64;
    const float* T2j = T2g + (size_t)j*K2*32;
    for (int t = 0; t < 15; ++t){
      const int col = 16*t + ii;
      v8f G = {};
      #pragma unroll
      for (int s = 0; s < 8; ++s){
        int kk = 4*s + kadd;
        v2f A; A.x = hls[ii][kk]; A.y = hls[ii][kk+1];
        v2f B = W2v[(s*2 + half)*240 + col];
        G = wmma4(A, B, G);
      }
      if (col < 227){
        float bias = bm2[col];
        #pragma unroll
        for (int rg=0; rg<8; ++rg){
          int m = 8*half + rg;                 // D row = pair index in tile
          float g  = G[rg] + bias;
          float gc = g * angs[m][8];           // * cm
          float* frow = fglob + (size_t)(i0 + m) * FROW;
          if (col < 129){
            atomicAdd(&frow[col], gc * scal[m][64 + col]);       // M0 (incl cm col)
          } else if (col < 194){
            int u = col - 129;
            if (u < 64){
              float a0=0.f, a1=0.f, a2=0.f;
              for (int k=0;k<K1;++k){
                float tv = T1j[k*64 + u];
                a0 += tv * B1s[m][k][0];
                a1 += tv * B1s[m][k][1];
                a2 += tv * B1s[m][k][2];
              }
              atomicAdd(&frow[129 + u*3 + 0], gc * a0);
              atomicAdd(&frow[129 + u*3 + 1], gc * a1);
              atomicAdd(&frow[129 + u*3 + 2], gc * a2);
            } else {                            // u == 64 -> A[1] row
              atomicAdd(&frow[129 + 192 + 0], gc * angs[m][0]);
              atomicAdd(&frow[129 + 192 + 1], gc * angs[m][1]);
              atomicAdd(&frow[129 + 192 + 2], gc * angs[m][2]);
            }
          } else {
            int u = col - 194;
            if (u < 32){
              float a0=0.f,a1=0.f,a2=0.f,a3=0.f,a4=0.f;
              for (int k=0;k<K2;++k){
                float tv = T2j[k*32 + u];
                a0 += tv * B2s[m][k][0];
                a1 += tv * B2s[m][k][1];
                a2 += tv * B2s[m][k][2];
                a3 += tv * B2s[m][k][3];
                a4 += tv * B2s[m][k][4];
              }
              atomicAdd(&frow[324 + u*5 + 0], gc * a0);
              atomicAdd(&frow[324 + u*5 + 1], gc * a1);
              atomicAdd(&frow[324 + u*5 + 2], gc * a2);
              atomicAdd(&frow[324 + u*5 + 3], gc * a3);
              atomicAdd(&frow[324 + u*5 + 4], gc * a4);
            } else {                            // u == 32 -> A[2] row
              atomicAdd(&frow[324 + 160 + 0], gc * angs[m][3]);
              atomicAdd(&frow[324 + 160 + 1], gc * angs[m][4]);
              atomicAdd(&frow[324 + 160 + 2], gc * angs[m][5]);
              atomicAdd(&frow[324 + 160 + 3], gc * angs[m][6]);
              atomicAdd(&frow[324 + 160 + 4], gc * angs[m][7]);
            }
          }
        }
      }
    }
  }
}

// ---------------------------------------------------------------------------
// Kernel 4: masked mean + output linears + coordinate update
// ---------------------------------------------------------------------------
__global__ __launch_bounds__(128) void eq_finalize(
    const float* __restrict__ fglob, const unsigned char* __restrict__ mask,
    const float* __restrict__ coord,
    const float* __restrict__ L0, const float* __restrict__ L1,
    const float* __restrict__ L2, const float* __restrict__ Lm,
    float* __restrict__ out)
{
  const int i = blockIdx.x;
  const int tid = threadIdx.x;
  __shared__ float scl;
  __shared__ float co[3][64];
  if (tid == 0){
    int c = 0;
    for (int j=0;j<NN;++j) c += (mask[j] != 0);
    float cnt = (mask[i] != 0) ? (float)c : 0.f;
    float denom = cnt + 1e-6f;
    float valid = (cnt > 1.f) ? 1.f : 0.f;
    scl = valid / denom;
  }
  __syncthreads();
  const float s = scl;
  const float* f = fglob + (size_t)i * FROW;

  { // o0 = f0[129] @ L0[129 x 128]
    float acc = 0.f;
    for (int u=0;u<129;++u) acc += f[u] * L0[u*128 + tid];
    out[(size_t)i*DFEAT + tid] = acc * s;
  }
  if (tid < 64){ // o1[v][b] = sum_u f1[u][b] * L1[u][v]
    const int v = tid;
    #pragma unroll
    for (int b=0;b<3;++b){
      float acc = 0.f;
      for (int u=0;u<65;++u) acc += f[129 + u*3 + b] * L1[u*64 + v];
      float o = acc * s;
      out[(size_t)i*DFEAT + 128 + v*3 + b] = o;
      co[b][v] = o * Lm[v];
    }
  }
  if (tid >= 64 && tid < 96){ // o2[v][b]
    const int v = tid - 64;
    #pragma unroll
    for (int b=0;b<5;++b){
      float acc = 0.f;
      for (int u=0;u<33;++u) acc += f[324 + u*5 + b] * L2[u*32 + v];
      out[(size_t)i*DFEAT + 320 + v*5 + b] = acc * s;
    }
  }
  __syncthreads();
  if (tid < 3){
    float acc = 0.f;
    for (int v=0;v<64;++v) acc += co[tid][v];
    out[(size_t)NN*DFEAT + (size_t)i*3 + tid] = coord[3*i + tid] + 1e-3f * acc;
  }
}

// ---------------------------------------------------------------------------
// Launch
// ---------------------------------------------------------------------------
extern "C" void kernel_launch(void* const* d_in, const int* in_sizes, int n_in,
                              void* d_out, int out_size, void* d_ws, size_t ws_size,
                              hipStream_t stream) {
  (void)in_sizes; (void)n_in; (void)out_size; (void)ws_size;
  const float*         x     = (const float*)d_in[0];
  const float*         coord = (const float*)d_in[1];
  const unsigned char* mask  = (const unsigned char*)d_in[2];   // numpy bool
  const float*         Wtp0  = (const float*)d_in[3];
  const float*         Wtp1  = (const float*)d_in[4];
  const float*         Wtp2  = (const float*)d_in[5];
  const float*         emb   = (const float*)d_in[6];
  const float*         Wm1   = (const float*)d_in[7];
  const float*         bm1   = (const float*)d_in[8];
  const float*         Wm2   = (const float*)d_in[9];
  const float*         bm2   = (const float*)d_in[10];
  const float*         L0    = (const float*)d_in[11];
  const float*         L1    = (const float*)d_in[12];
  const float*         L2    = (const float*)d_in[13];
  const float*         Lm    = (const float*)d_in[14];
  float* out = (float*)d_out;

  float* ws    = (float*)d_ws;
  float* fglob = ws;                               // NN*FROW
  float* T0    = fglob + (size_t)NN*FROW;          // NN*K0*128
  float* T1    = T0 + (size_t)NN*K0*128;           // NN*K1*64
  float* T2    = T1 + (size_t)NN*K1*64;            // NN*K2*32
  float* CGP0  = T2 + (size_t)NN*K2*32;            // 9*5
  float* CGP1  = CGP0 + K0*5;                      // 20*5*3
  float* CGP2  = CGP1 + K1*5*3;                    // 22*5*5
  float* W1p   = CGP2 + K2*5*5;                    // W1P_ELEMS
  float* W2p   = W1p + W1P_ELEMS;                  // W2P_ELEMS

  hipMemsetAsync(fglob, 0, (size_t)NN*FROW*sizeof(float), stream);
  eq_init_cg<<<1, 32, 0, stream>>>(CGP0, CGP1, CGP2);
  eq_pack_w<<<16, 256, 0, stream>>>(Wm1, Wm2, W1p, W2p);
  eq_stage_t<<<NN, 128, 0, stream>>>(x, Wtp0, Wtp1, Wtp2, T0, T1, T2);
  dim3 pg(NN/16, 64);   // 24 i-tiles x 64 j-chunks (6 j each)
  eq_pair_kernel<<<pg, 32, 0, stream>>>(coord, mask, emb, bm1, bm2, W1p, W2p,
                                        T0, T1, T2, CGP0, CGP1, CGP2, fglob, 6);
  eq_finalize<<<NN, 128, 0, stream>>>(fglob, mask, coord, L0, L1, L2, Lm, out);
}